// OptimizedFractalDenoise1D_18777597018854
// MI455X (gfx1250) — compile-verified
//
#include <hip/hip_runtime.h>

// ---------------------------------------------------------------------------
// FractalDenoise1D for MI455X (gfx1250, wave32).
//
// Key simplification: the reference's overlap-add (R=5, STRIDE=2) covers every
// position >=1 time and divides by the coverage count -> it is an identity.
// Each iteration is therefore:
//   local = box5(x), trend = box11(x) (reflect pad)
//   thr   = 3.5 * max(std(x - local, ddof=1, per row), 1e-6)
//   y     = 0.4*local + 0.6*trend + 0.85 * (|r|>thr ? 0.35r : r),  r = x-local
//
// Bandwidth-bound (67 MB tensor, ~0.03 FLOP/byte); working set fits in 192 MB
// L2, so the 3-kernel/iteration pipeline keeps all re-reads on-chip.
//
// Box filters run as banded-matrix GEMMs on the precision-safe f32 WMMA:
// D(16x16) = W(16x28) * X(28x16), chained as 7x K=4 V_WMMA_F32_16X16X4_F32
// per filter per wave per 256 outputs (verified lowering in round 1).
// Tile staging uses GLOBAL_LOAD_ASYNC_TO_LDS_B32 (+ s_wait_asynccnt) on
// interior tiles; reflect-indexed VGPR loads on edge tiles.
//
// d_ws layout (floats): [0,256) stats iter1, [256,512) stats iter2,
// [512,640) thr1, [640,768) thr2, [1024, 1024+16M) y1 intermediate.
// Requires ws_size >= 4 KB + 64 MiB.
// ---------------------------------------------------------------------------

typedef float v2f __attribute__((ext_vector_type(2)));
typedef float v8f __attribute__((ext_vector_type(8)));

#define L_LEN   131072
#define ROWS    128            // B*C = 8*16
#define TILE    2048
#define NTILES  (L_LEN / TILE) // 64
#define HALO_L  5
#define LDS_Y   (TILE + HALO_L + 7)  // left halo 5, right halo 7 (K padded to 28)

#if defined(__HIP_DEVICE_COMPILE__) && defined(__gfx1250__) && \
    __has_builtin(__builtin_amdgcn_global_load_async_to_lds_b32) && \
    __has_builtin(__builtin_amdgcn_s_wait_asynccnt)
#define USE_ASYNC_LDS 1
#else
#define USE_ASYNC_LDS 0
#endif

__device__ __forceinline__ int reflect_idx(int i, int L) {
    // jnp.pad 'reflect': -1 -> 1, L -> L-2. Halo <= 22 so one reflection is enough.
    if (i < 0)  i = -i;
    if (i >= L) i = 2 * L - 2 - i;
    return i;
}

// Stage lds[0..n) = x[rowBase + g0 + i] (reflect-clamped). On interior tiles the
// mapping is linear, so use direct global->LDS async copies (no VGPR staging).
__device__ __forceinline__ void stage_row_tile(const float* __restrict__ x,
                                               float* lds, int n,
                                               size_t rowBase, int g0,
                                               bool interior) {
#if USE_ASYNC_LDS
    if (interior) {
        const float* src = x + rowBase + g0;
        for (int i = threadIdx.x; i < n; i += 256) {
            __builtin_amdgcn_global_load_async_to_lds_b32(
                (__attribute__((address_space(1))) int*)(src + i),
                (__attribute__((address_space(3))) int*)(lds + i),
                /*offset=*/0, /*cpol=*/0);
        }
        __builtin_amdgcn_s_wait_asynccnt(0);
        __syncthreads();
        return;
    }
#endif
    for (int i = threadIdx.x; i < n; i += 256)
        lds[i] = x[rowBase + reflect_idx(g0 + i, L_LEN)];
    __syncthreads();
}

// ---- zero the stats scratch every call (no cross-call state) ---------------
__global__ void init_kernel(float* wsf) {
    int i = threadIdx.x + blockIdx.x * blockDim.x;
    if (i < 768) wsf[i] = 0.0f;
}

// ---- per-row sum / sum-of-squares of residual = x - box5(x) ----------------
__global__ __launch_bounds__(256) void stats_kernel(const float* __restrict__ x,
                                                    float* __restrict__ stats) {
    __shared__ float lds[TILE + 4];            // halo +/-2 for box5
    __shared__ float ps[8], ps2[8];
    const int row   = blockIdx.y;
    const int tile0 = blockIdx.x * TILE;
    const size_t rowBase = (size_t)row * L_LEN;
    const bool interior = (blockIdx.x > 0) && (blockIdx.x < NTILES - 1);

    stage_row_tile(x, lds, TILE + 4, rowBase, tile0 - 2, interior);

    float s = 0.0f, s2 = 0.0f;
    const int p0 = threadIdx.x * 8 + 2;        // 8 consecutive elems per thread
    #pragma unroll
    for (int e = 0; e < 8; ++e) {
        const int c  = p0 + e;
        const float loc = 0.2f * (lds[c-2] + lds[c-1] + lds[c] + lds[c+1] + lds[c+2]);
        const float r   = lds[c] - loc;
        s += r; s2 += r * r;
    }
    // wave32 reduction
    #pragma unroll
    for (int off = 16; off > 0; off >>= 1) {
        s  += __shfl_down(s,  off, 32);
        s2 += __shfl_down(s2, off, 32);
    }
    const int wave = threadIdx.x >> 5, lane = threadIdx.x & 31;
    if (lane == 0) { ps[wave] = s; ps2[wave] = s2; }
    __syncthreads();
    if (threadIdx.x == 0) {
        float ts = 0.0f, ts2 = 0.0f;
        #pragma unroll
        for (int i = 0; i < 8; ++i) { ts += ps[i]; ts2 += ps2[i]; }
        atomicAdd(&stats[row * 2 + 0], ts);
        atomicAdd(&stats[row * 2 + 1], ts2);
    }
}

// ---- per-row spike threshold ----------------------------------------------
__global__ void finalize_kernel(const float* __restrict__ stats,
                                float* __restrict__ thr) {
    const int r = threadIdx.x;
    if (r < ROWS) {
        const float s  = stats[r * 2 + 0];
        const float s2 = stats[r * 2 + 1];
        const float Lf = (float)L_LEN;
        float var = (s2 - s * s / Lf) / (Lf - 1.0f);   // ddof = 1
        var = fmaxf(var, 0.0f);
        thr[r] = 3.5f * fmaxf(sqrtf(var), 1e-6f);
    }
}

// ---- fused box5 + box11 via f32 WMMA, spike damp, combine ------------------
// output p = base + m + 16*n; X[k][n] = x[base + 16n + k - 5], k in [0,28)
// W_local[m][k] = 1/5  for k in [m+3, m+7]
// W_trend[m][k] = 1/11 for k in [m,   m+10]
// A layout (ISA 7.12.2, f32 A 16x4): lane l -> M = l&15, K = c + 2*(l>>4).
// B 4x16 assumed to mirror A. C/D: reg r, lane l -> M = r + 8*(l>>4), N = l&15.
__global__ __launch_bounds__(256) void apply_kernel(const float* __restrict__ x,
                                                    const float* __restrict__ thr,
                                                    float* __restrict__ y) {
    __shared__ float lds[LDS_Y];
    const int row   = blockIdx.y;
    const int tile0 = blockIdx.x * TILE;
    const size_t rowBase = (size_t)row * L_LEN;
    const bool interior = (blockIdx.x > 0) && (blockIdx.x < NTILES - 1);

    stage_row_tile(x, lds, LDS_Y, rowBase, tile0 - HALO_L, interior);

    const float t    = thr[row];
    const int wave   = threadIdx.x >> 5;
    const int lane   = threadIdx.x & 31;
    const int m      = lane & 15;      // A row / B col (mirrored layout)
    const int kh     = lane >> 4;      // K half-select
    const int wbase  = wave * 256;     // this wave's 256-output chunk (LDS-rel)
    const int bcol   = wbase + 16 * m; // + k  (the -5 tile halo cancels +(-5) of X)

    v8f accL = {};  // box5
    v8f accT = {};  // box11
    #pragma unroll
    for (int s = 0; s < 7; ++s) {
        const int k0 = 4 * s + 2 * kh;
        const int k1 = k0 + 1;
        v2f aL, aT, b;
        // unsigned-wrap range checks: 1 cmp + 1 cndmask per weight
        aL.x = ((unsigned)(k0 - (m + 3)) <= 4u ) ? 0.2f           : 0.0f;
        aL.y = ((unsigned)(k1 - (m + 3)) <= 4u ) ? 0.2f           : 0.0f;
        aT.x = ((unsigned)(k0 -  m     ) <= 10u) ? (1.0f / 11.0f) : 0.0f;
        aT.y = ((unsigned)(k1 -  m     ) <= 10u) ? (1.0f / 11.0f) : 0.0f;
        b.x  = lds[bcol + k0];
        b.y  = lds[bcol + k1];
        // (neg_a, A, neg_b, B, c_mod, C, reuse_a, reuse_b)
        accL = __builtin_amdgcn_wmma_f32_16x16x4_f32(false, aL, false, b,
                                                     (short)0, accL, false, false);
        accT = __builtin_amdgcn_wmma_f32_16x16x4_f32(false, aT, false, b,
                                                     (short)0, accT, false, false);
    }

    // D readout: reg r -> M = r + 8*kh (consecutive addresses across r)
    const int    pl    = wbase + HALO_L + 8 * kh + 16 * m;             // lds idx of r=0
    const size_t gbase = rowBase + tile0 + wbase + 8 * kh + 16 * m;    // global idx of r=0
    float o[8];
    #pragma unroll
    for (int r = 0; r < 8; ++r) {
        const float xc  = lds[pl + r];
        const float loc = accL[r];
        const float trd = accT[r];
        float res = xc - loc;
        res = (fabsf(res) > t) ? res * 0.35f : res;
        o[r] = 0.4f * loc + 0.6f * trd + 0.85f * res;
    }
    *reinterpret_cast<float4*>(&y[gbase])     = make_float4(o[0], o[1], o[2], o[3]);
    *reinterpret_cast<float4*>(&y[gbase + 4]) = make_float4(o[4], o[5], o[6], o[7]);
}

// ---------------------------------------------------------------------------
extern "C" void kernel_launch(void* const* d_in, const int* in_sizes, int n_in,
                              void* d_out, int out_size, void* d_ws, size_t ws_size,
                              hipStream_t stream) {
    (void)in_sizes; (void)n_in; (void)out_size; (void)ws_size;
    const float* x   = (const float*)d_in[0];
    float*       out = (float*)d_out;
    float*       wsf = (float*)d_ws;

    float* stats1 = wsf;            // 256 floats
    float* stats2 = wsf + 256;      // 256 floats
    float* thr1   = wsf + 512;      // 128 floats
    float* thr2   = wsf + 640;      // 128 floats
    float* y1     = wsf + 1024;     // 16M floats (iteration-1 output)

    const dim3 grid(NTILES, ROWS);
    const dim3 blk(256);

    init_kernel<<<dim3(3), blk, 0, stream>>>(wsf);

    // iteration 1: x -> y1
    stats_kernel   <<<grid, blk, 0, stream>>>(x, stats1);
    finalize_kernel<<<1, 128, 0, stream>>>(stats1, thr1);
    apply_kernel   <<<grid, blk, 0, stream>>>(x, thr1, y1);

    // iteration 2: y1 -> out
    stats_kernel   <<<grid, blk, 0, stream>>>(y1, stats2);
    finalize_kernel<<<1, 128, 0, stream>>>(stats2, thr2);
    apply_kernel   <<<grid, blk, 0, stream>>>(y1, thr2, out);
}